// GraphSAGE_59004260713169
// MI455X (gfx1250) — compile-verified
//
#include <hip/hip_runtime.h>

typedef __attribute__((ext_vector_type(2))) float v2f;
typedef __attribute__((ext_vector_type(8))) float v8f;

#define F_DIM 128
#define LDA   132   // 128 + 4 dword pad -> conflict-free LDS rows

// ---------------------------------------------------------------- zero fill
__global__ void zero_kernel(float* __restrict__ p, long long n) {
  long long i = (long long)blockIdx.x * blockDim.x + threadIdx.x;
  long long stride = (long long)gridDim.x * blockDim.x;
  for (; i < n; i += stride) p[i] = 0.0f;
}

// ------------------------------------------------- scatter-add (mean numer.)
// One wave (32 lanes) per edge; lane L handles features [4L, 4L+4).
__global__ void scatter_kernel(const float* __restrict__ feat,
                               const long long* __restrict__ eidx,
                               float* __restrict__ agg,
                               float* __restrict__ deg,   // null on 2nd pass
                               int E) {
  long long t = (long long)blockIdx.x * blockDim.x + threadIdx.x;
  int lane = (int)(t & 31);
  int e = (int)(t >> 5);
  if (e >= E) return;
  if (e + 1024 < E) __builtin_prefetch(eidx + e + 1024, 0, 1);  // global_prefetch_b8
  int s = (int)eidx[e];
  int d = (int)eidx[(long long)E + e];
  const float4 v = *(const float4*)(feat + (unsigned)s * F_DIM + lane * 4);
  float* ap = agg + (unsigned)d * F_DIM + lane * 4;
  atomicAdd(ap + 0, v.x);
  atomicAdd(ap + 1, v.y);
  atomicAdd(ap + 2, v.z);
  atomicAdd(ap + 3, v.w);
  if (deg != nullptr && lane == 0) atomicAdd(deg + d, 1.0f);
}

// ---------------------------------------------------- fused SAGE layer GEMM
// out[row, col] = act( (agg[row]/max(deg,1)) @ Wl + xroot[row] @ Wr + bias )
// Block: 256 threads = 8 waves, 16 node-rows per block.
// Wave w computes the 16x16 output tile at columns [16w, 16w+16) via
// V_WMMA_F32_16X16X4_F32 (K-loop over 128 in steps of 4, 2 WMMA per step).
// NOTE on padding (NCOLS=40 case): lane L's B-fragment values only feed
// output column L, and lanes with col>=NCOLS are never stored, so clamped
// (in-range) garbage loads need no zero-masking -> branchless inner loop.
template <int NCOLS, int NTILES, bool RELU>
__global__ void __launch_bounds__(256)
sage_gemm_wmma(const float* __restrict__ agg, const float* __restrict__ deg,
               const float* __restrict__ xroot,
               const float* __restrict__ Wl, const float* __restrict__ Wr,
               const float* __restrict__ bias, float* __restrict__ out,
               int nrows) {
  __shared__ float sMean[16 * LDA];
  __shared__ float sX[16 * LDA];
  __shared__ float sInv[16];

  const int tid = threadIdx.x;
  const int r0 = blockIdx.x * 16;

  if (tid < 16) {
    int row = r0 + tid;
    if (row > nrows - 1) row = nrows - 1;
    sInv[tid] = 1.0f / fmaxf(deg[row], 1.0f);
  }
  __syncthreads();

  // Stage 16x128 tiles of mean and root features into LDS (8 elems/thread).
  for (int i = tid; i < 16 * F_DIM; i += 256) {
    int r = i >> 7;
    int k = i & (F_DIM - 1);
    int row = r0 + r;
    if (row > nrows - 1) row = nrows - 1;
    sMean[r * LDA + k] = agg[(unsigned)row * F_DIM + k] * sInv[r];
    sX[r * LDA + k] = xroot[(unsigned)row * F_DIM + k];
  }
  __syncthreads();

  const int wave = tid >> 5;
  const int lane = tid & 31;
  if (wave < NTILES) {                   // wave-uniform branch, EXEC stays full
    const int col_base = wave * 16;
    const int rl = lane & 15;            // A: M row / B: N col within tile
    const int koff = (lane >> 4) * 2;    // lane-half K split {0,1} vs {2,3}
    const int col = col_base + rl;
    const bool colok = ((NCOLS & 15) == 0) ? true : (col < NCOLS);
    // Clamped (always in-range) column for unguarded loads.
    const int colc = ((NCOLS & 15) == 0) ? col : (colok ? col : (NCOLS - 1));

    v8f c = {};
#pragma unroll 2
    for (int k = 0; k < F_DIM; k += 4) {
      v2f am, ax, bl, br;
      const float* mrow = &sMean[rl * LDA + k + koff];
      const float* xrow = &sX[rl * LDA + k + koff];
      am.x = mrow[0]; am.y = mrow[1];
      ax.x = xrow[0]; ax.y = xrow[1];
      const int kk = k + koff;
      bl.x = Wl[kk * NCOLS + colc];
      bl.y = Wl[(kk + 1) * NCOLS + colc];
      br.x = Wr[kk * NCOLS + colc];
      br.y = Wr[(kk + 1) * NCOLS + colc];
      c = __builtin_amdgcn_wmma_f32_16x16x4_f32(false, am, false, bl,
                                                (short)0, c, false, false);
      c = __builtin_amdgcn_wmma_f32_16x16x4_f32(false, ax, false, br,
                                                (short)0, c, false, false);
    }

    const float bv = bias[colc];
    const int rhalf = (lane >> 4) << 3;  // lanes 16-31 hold M = r + 8
    if (r0 + 16 <= nrows) {
      // Full tile (always taken when nrows % 16 == 0): no per-row guards.
      // For NCOLS%16!=0, colok is loop-invariant -> one EXEC mask, 8 stores.
#pragma unroll
      for (int r = 0; r < 8; ++r) {
        if (colok) {
          int row = r0 + r + rhalf;
          float v = c[r] + bv;
          if (RELU) v = fmaxf(v, 0.0f);
          out[(unsigned)row * NCOLS + col] = v;
        }
      }
    } else {
#pragma unroll
      for (int r = 0; r < 8; ++r) {
        int row = r0 + r + rhalf;
        if ((row < nrows) && colok) {
          float v = c[r] + bv;
          if (RELU) v = fmaxf(v, 0.0f);
          out[(unsigned)row * NCOLS + col] = v;
        }
      }
    }
  }
}

extern "C" void kernel_launch(void* const* d_in, const int* in_sizes, int n_in,
                              void* d_out, int out_size, void* d_ws, size_t ws_size,
                              hipStream_t stream) {
  const float* x = (const float*)d_in[0];
  const long long* eidx = (const long long*)d_in[1];
  const float* Wl1 = (const float*)d_in[2];
  const float* Wr1 = (const float*)d_in[3];
  const float* b1 = (const float*)d_in[4];
  const float* Wl2 = (const float*)d_in[5];
  const float* Wr2 = (const float*)d_in[6];
  const float* b2 = (const float*)d_in[7];
  float* out = (float*)d_out;

  const int N = in_sizes[0] / F_DIM;
  const int E = in_sizes[1] / 2;

  // Workspace layout (floats): [deg: N | agg: N*128 | h: N*128]
  float* deg = (float*)d_ws;
  float* agg = deg + N;
  float* h = agg + (size_t)N * F_DIM;

  const unsigned zgrid = 2048;
  const unsigned sblocks = (unsigned)(((long long)E * 32 + 255) / 256);
  const unsigned gblocks = (unsigned)((N + 15) / 16);

  // ---- Layer 1 ----
  zero_kernel<<<zgrid, 256, 0, stream>>>(deg, (long long)N * (F_DIM + 1));
  scatter_kernel<<<sblocks, 256, 0, stream>>>(x, eidx, agg, deg, E);
  sage_gemm_wmma<128, 8, true>
      <<<gblocks, 256, 0, stream>>>(agg, deg, x, Wl1, Wr1, b1, h, N);

  // ---- Layer 2 ----
  zero_kernel<<<zgrid, 256, 0, stream>>>(agg, (long long)N * F_DIM);
  scatter_kernel<<<sblocks, 256, 0, stream>>>(h, eidx, agg, nullptr, E);
  sage_gemm_wmma<40, 3, false>
      <<<gblocks, 256, 0, stream>>>(agg, deg, h, Wl2, Wr2, b2, out, N);
}